// PrologGrounder_29394756174279
// MI455X (gfx1250) — compile-verified
//
#include <hip/hip_runtime.h>
#include <hip/hip_bf16.h>
#include <stdint.h>

// ---- problem constants (match reference) ----
#define PADV      0
#define CONST_NO  10000
#define Bq        8
#define Sq        64
#define Gq        8
#define Fq        200000
#define Rq        200
#define BMAXq     3
#define KFq       64
#define KRq       32

// ---- tiling for the fact scan ----
#define TILE_FACTS        2048                       // facts per LDS tile
#define TILE_BYTES        (TILE_FACTS * 12)          // 24576 B
#define COPIES_PER_THREAD (TILE_BYTES / (256 * 16))  // 6 x b128 per thread
#define FACTS_PER_THREAD  (TILE_FACTS / 256)         // 8
#define NWAVES            8                          // 256 threads / wave32
#define TOTAL_FACT_BYTES  ((long)Fq * 12)            // 2,400,000 (16B multiple)
#define NUM_TILES         ((Fq + TILE_FACTS - 1) / TILE_FACTS)  // 98

#define WS_STRIDE 128   // ints per query in workspace: [nf, nr, fidx[64], ridx[32], pad]

// ---- CDNA5 async global->LDS path (guarded: falls back to plain copies) ----
#if defined(__has_builtin)
#if __has_builtin(__builtin_amdgcn_global_load_async_to_lds_b128) && \
    __has_builtin(__builtin_amdgcn_s_wait_asynccnt)
#define USE_ASYNC_LDS 1
#endif
#endif
#ifndef USE_ASYNC_LDS
#define USE_ASYNC_LDS 0
#endif

// The async builtins take pointers to 16B int vectors in AS1 (global) / AS3 (LDS).
typedef int async_v4i __attribute__((vector_size(16)));
#define AS1V4(p) ((__attribute__((address_space(1))) async_v4i*)(p))
#define AS3V4(p) ((__attribute__((address_space(3))) async_v4i*)(p))

// =====================================================================
// Kernel 1: one block (8 wave32) per (b,s) query. Scan 200k facts via
// double-buffered async LDS tiles; ordered compaction of the first 64
// matching fact indices using shfl-based wave scans (2 barriers/tile).
// Rules (R=200) matched serially by thread 0. Results -> workspace.
// =====================================================================
__global__ __launch_bounds__(256) void grounder_match_kernel(
    const int* __restrict__ goals,   // (B,S,G,3) int32
    const int* __restrict__ facts,   // (F,3) int32
    const int* __restrict__ heads,   // (R,3) int32
    int* __restrict__ ws)
{
  const int q    = blockIdx.x;      // b*S + s
  const int t    = threadIdx.x;     // 0..255
  const int lane = t & 31;
  const int wid  = t >> 5;          // wave id 0..7

  __shared__ __align__(16) int ldsFacts[2][TILE_FACTS * 3];  // 2 x 24KB
  __shared__ int sWaveTot[NWAVES];
  __shared__ int sFidx[KFq];
  __shared__ int sRidx[KRq];
  __shared__ int sQ[3];
  __shared__ int sNr;

  if (t == 0) {
    const int qp  = goals[(q * Gq + 0) * 3 + 0];
    const int qa0 = goals[(q * Gq + 0) * 3 + 1];
    const int qa1 = goals[(q * Gq + 0) * 3 + 2];
    sQ[0] = qp; sQ[1] = qa0; sQ[2] = qa1;
    const bool act = (qp != PADV);
    int nr = 0;
    for (int r = 0; r < Rq && nr < KRq; ++r) {
      const int hp = heads[r * 3 + 0];
      const int h0 = heads[r * 3 + 1];
      const int h1 = heads[r * 3 + 2];
      const bool ok = act && (qp == hp) &&
                      ((h0 > CONST_NO) || (qa0 > CONST_NO) || (qa0 == h0)) &&
                      ((h1 > CONST_NO) || (qa1 > CONST_NO) || (qa1 == h1));
      if (ok) sRidx[nr++] = r;
    }
    sNr = nr;
  }
  __syncthreads();

  const int  qp     = sQ[0];
  const int  qa0    = sQ[1];
  const int  qa1    = sQ[2];
  const bool active = (qp != PADV);

  const char* factBytes = (const char*)facts;

  // Issue one tile's worth of copies. Each thread always issues exactly
  // COPIES_PER_THREAD b128 ops (tail addresses clamped in-bounds) so every
  // wave's ASYNCcnt rises by the same fixed amount per tile.
  auto issueTile = [&](int buf, int tile) {
    const long gbase = (long)tile * TILE_BYTES;
#pragma unroll
    for (int jdx = 0; jdx < COPIES_PER_THREAD; ++jdx) {
      const long lb = (long)jdx * 4096 + (long)t * 16;
      long gb = gbase + lb;
      if (gb > TOTAL_FACT_BYTES - 16) gb = TOTAL_FACT_BYTES - 16;  // clamp tail
#if USE_ASYNC_LDS
      __builtin_amdgcn_global_load_async_to_lds_b128(
          AS1V4(factBytes + gb),
          AS3V4((char*)&ldsFacts[buf][0] + lb),
          0, 0);
#else
      *(int4*)((char*)&ldsFacts[buf][0] + lb) = *(const int4*)(factBytes + gb);
#endif
    }
  };

  int matchTotal = 0;  // uniform across the block (every thread computes it)

  if (active) {
    issueTile(0, 0);  // prime the pipeline

    for (int tile = 0; tile < NUM_TILES; ++tile) {
      const int  cur  = tile & 1;
      const bool pref = (tile + 1) < NUM_TILES;
      if (pref) issueTile(cur ^ 1, tile + 1);   // prefetch next tile
#if USE_ASYNC_LDS
      if (pref) __builtin_amdgcn_s_wait_asynccnt(COPIES_PER_THREAD); // tile done, prefetch in flight
      else      __builtin_amdgcn_s_wait_asynccnt(0);
#endif
      __syncthreads();  // all waves waited -> whole tile resident in LDS
                        // (also orders prev tile's sWaveTot reads vs rewrites)

      const int base = tile * TILE_FACTS;
      int m[FACTS_PER_THREAD];
      int cnt = 0;
#pragma unroll
      for (int i = 0; i < FACTS_PER_THREAD; ++i) {
        const int l = t * FACTS_PER_THREAD + i;   // contiguous -> ascending order
        const int f = base + l;
        int ok = 0;
        if (f < Fq) {
          const int fp = ldsFacts[cur][l * 3 + 0];
          const int fa = ldsFacts[cur][l * 3 + 1];
          const int fb = ldsFacts[cur][l * 3 + 2];
          ok = (qp == fp) && ((qa0 > CONST_NO) || (qa0 == fa)) &&
               ((qa1 > CONST_NO) || (qa1 == fb));
        }
        m[i] = ok;
        cnt += ok;
      }

      // intra-wave inclusive scan of per-thread counts (no barriers, wave32)
      int incl = cnt;
#pragma unroll
      for (int off = 1; off < 32; off <<= 1) {
        const int v = __shfl_up(incl, off, 32);
        if (lane >= off) incl += v;
      }
      if (lane == 31) sWaveTot[wid] = incl;
      __syncthreads();

      // combine wave totals: exclusive base for this wave + tile total
      int wbase = 0, tot = 0;
#pragma unroll
      for (int w = 0; w < NWAVES; ++w) {
        const int wt = sWaveTot[w];
        if (w < wid) wbase += wt;
        tot += wt;
      }

      int pos = matchTotal + wbase + (incl - cnt);  // ordered global rank
#pragma unroll
      for (int i = 0; i < FACTS_PER_THREAD; ++i) {
        if (m[i]) {
          if (pos < KFq) sFidx[pos] = base + t * FACTS_PER_THREAD + i;
          ++pos;
        }
      }

      matchTotal += tot;              // uniform -> uniform early exit
      if (matchTotal >= KFq) break;
    }
  }

  __syncthreads();  // sFidx stores visible to all before writing out

  const int nf = (matchTotal > KFq) ? KFq : matchTotal;
  int* wq = ws + (long)q * WS_STRIDE;
  if (t == 0) { wq[0] = nf; wq[1] = sNr; }
  if (t < KFq) wq[2 + t]       = (t < nf)  ? sFidx[t] : 0;
  if (t < KRq) wq[2 + KFq + t] = (t < sNr) ? sRidx[t] : 0;
}

// =====================================================================
// Kernel 2: one block per batch b. Prefix-sum success counts over s,
// map each of the S=64 output slots to its (s, rank), substitute, emit.
// =====================================================================
__global__ __launch_bounds__(64) void grounder_emit_kernel(
    const int*   __restrict__ goals,        // (B,S,G,3)
    const float* __restrict__ state_scores, // (B,S)
    const int*   __restrict__ facts,        // (F,3)
    const float* __restrict__ fact_scores,  // (F,)
    const int*   __restrict__ heads,        // (R,3)
    const int*   __restrict__ bodies,       // (R,BMAX,3)
    const int*   __restrict__ rule_lens,    // (R,)
    const float* __restrict__ rule_scores,  // (R,)
    const int*   __restrict__ ws,
    float*       __restrict__ out)          // goals(B,S,G,3) then scores(B,S)
{
  const int b = blockIdx.x;
  const int j = threadIdx.x;   // output slot 0..63

  __shared__ int pref[Sq];     // inclusive prefix of (nf+nr) over s
  {
    const int* wq = ws + (long)(b * Sq + j) * WS_STRIDE;
    pref[j] = wq[0] + wq[1];
  }
  __syncthreads();
  for (int off = 1; off < Sq; off <<= 1) {
    const int v = (j >= off) ? pref[j - off] : 0;
    __syncthreads();
    pref[j] += v;
    __syncthreads();
  }
  const int total = pref[Sq - 1];

  int og[Gq * 3];
#pragma unroll
  for (int i = 0; i < Gq * 3; ++i) og[i] = PADV;
  float sc = 0.0f;

  if (j < total) {
    int s = 0;
    while (pref[s] <= j) ++s;                 // source state index
    const int* wq = ws + (long)(b * Sq + s) * WS_STRIDE;
    const int nf = wq[0];
    const int nr = wq[1];
    const int r  = j - (pref[s] - (nf + nr)); // rank within this s (facts first)
    const int* qg = goals + (long)(b * Sq + s) * Gq * 3;
    const int qa0 = qg[1];
    const int qa1 = qg[2];
    const float ss = state_scores[b * Sq + s];

    if (r < nf) {
      // ---- fact expansion: goal0 -> PAD, goals 1..7 = remaining w/ var subst
      const int f   = wq[2 + r];
      const int fb0 = facts[f * 3 + 1];
      const int fb1 = facts[f * 3 + 2];
      for (int g = 1; g < Gq; ++g)
        for (int c = 0; c < 3; ++c) {
          int x = qg[g * 3 + c];
          if (qa0 > CONST_NO && x == qa0) x = fb0;
          if (qa1 > CONST_NO && x == qa1) x = fb1;
          og[g * 3 + c] = x;
        }
      sc = ss * fact_scores[f];
    } else {
      // ---- rule expansion: goals 0..2 = body (subst + len mask), 3..7 = rem 1..5
      const int rr = wq[2 + KFq + (r - nf)];
      const int h0 = heads[rr * 3 + 1];
      const int h1 = heads[rr * 3 + 2];
      const int rl = rule_lens[rr];
      for (int g = 0; g < BMAXq; ++g) {
        if (g < rl) {
          for (int c = 0; c < 3; ++c) {
            int x = bodies[(rr * BMAXq + g) * 3 + c];
            if (h0 > CONST_NO && x == h0) x = qa0;
            if (h1 > CONST_NO && x == h1) x = qa1;
            og[g * 3 + c] = x;
          }
        }  // else: stays PAD
      }
      for (int g = BMAXq; g < Gq; ++g) {
        const int srcg = g - BMAXq + 1;  // remaining goals 1..5 (no subst)
        for (int c = 0; c < 3; ++c) og[g * 3 + c] = qg[srcg * 3 + c];
      }
      sc = ss * rule_scores[rr];
    }
  }

  // write goals (cast to float) + score; fully initializes poisoned d_out
  for (int g = 0; g < Gq; ++g)
    for (int c = 0; c < 3; ++c)
      out[(((long)b * Sq + j) * Gq + g) * 3 + c] = (float)og[g * 3 + c];
  out[(long)Bq * Sq * Gq * 3 + b * Sq + j] = sc;
}

// =====================================================================
extern "C" void kernel_launch(void* const* d_in, const int* in_sizes, int n_in,
                              void* d_out, int out_size, void* d_ws, size_t ws_size,
                              hipStream_t stream)
{
  (void)in_sizes; (void)n_in; (void)out_size; (void)ws_size;

  const int*   goals        = (const int*)  d_in[0];  // proof_goals
  const float* state_scores = (const float*)d_in[1];
  const int*   facts        = (const int*)  d_in[2];
  const float* fact_scores  = (const float*)d_in[3];
  const int*   heads        = (const int*)  d_in[4];
  const int*   bodies       = (const int*)  d_in[5];
  const int*   rule_lens    = (const int*)  d_in[6];
  const float* rule_scores  = (const float*)d_in[7];

  int*   ws  = (int*)  d_ws;   // needs 512 * 128 * 4 = 256 KB
  float* out = (float*)d_out;

  grounder_match_kernel<<<Bq * Sq, 256, 0, stream>>>(goals, facts, heads, ws);
  grounder_emit_kernel<<<Bq, 64, 0, stream>>>(goals, state_scores, facts, fact_scores,
                                              heads, bodies, rule_lens, rule_scores,
                                              ws, out);
}